// RelPosMultiheadAttention_10127532884371
// MI455X (gfx1250) — compile-verified
//
#include <hip/hip_runtime.h>
#include <math.h>

#define EMBED   768
#define NH      12
#define HD      64
#define MAXREL  50
#define NB      4
#define SEQ     1024
#define NQKV    (3*EMBED)

typedef __attribute__((ext_vector_type(2))) float v2f;
typedef __attribute__((ext_vector_type(8))) float v8f;

static __device__ __forceinline__ v8f wmma_f32(v2f a, v2f b, v8f c) {
  // 8 args: (neg_a, A, neg_b, B, c_mod, C, reuse_a, reuse_b)
  return __builtin_amdgcn_wmma_f32_16x16x4_f32(false, a, false, b, (short)0, c, false, false);
}

// ---------------------------------------------------------------------------
// dt[b] = nanmedian(patch[b,1:]-patch[b,:-1]); fallback 1.0 (exact rank select)
// ---------------------------------------------------------------------------
__global__ __launch_bounds__(1024)
void median_dt_kernel(const float* __restrict__ patch, float* __restrict__ dt) {
  __shared__ float sd[SEQ - 1];
  __shared__ int   s_cnt;
  __shared__ float s_med;
  const int b = blockIdx.x;
  const int t = threadIdx.x;
  if (t == 0) { s_cnt = 0; s_med = 0.0f; }
  __syncthreads();
  float di = 0.0f; bool valid = false;
  if (t < SEQ - 1) {
    di = patch[b*SEQ + t + 1] - patch[b*SEQ + t];
    sd[t] = di;
    valid = !__builtin_isnan(di);
    if (valid) atomicAdd(&s_cnt, 1);
  }
  __syncthreads();
  const int cnt = s_cnt;
  if (valid && cnt > 0) {
    int rank = 0;
    for (int j = 0; j < SEQ - 1; ++j) {
      float dj = sd[j];                       // NaN compares false -> skipped
      rank += ((dj < di) || (dj == di && j < t)) ? 1 : 0;
    }
    if (cnt & 1) {
      if (rank == cnt / 2) s_med = di;        // unique writer
    } else {
      if (rank == cnt/2 - 1 || rank == cnt/2) atomicAdd(&s_med, 0.5f * di);
    }
  }
  __syncthreads();
  if (t == 0) {
    float m = s_med;
    bool ok = (!__builtin_isnan(m)) && (!__builtin_isinf(m)) && (m > 0.0f);
    dt[b] = ok ? m : 1.0f;
  }
}

// ---------------------------------------------------------------------------
// Wave-level 64x64 f32-WMMA GEMM core: acc[mt][nt] += A[64xK] * W[64xK]^T
// 16 WMMAs per 8 fragment loads (4x operand reuse vs 16x64 tiling).
// ---------------------------------------------------------------------------
__device__ __forceinline__ void wave_gemm_64x64(
    const float* __restrict__ A, const float* __restrict__ W,
    int m0, int n0, int kdim, int c, int hh, v8f acc[4][4]) {
  const float* ar0 = A + (size_t)(m0 +  0 + c) * kdim;
  const float* ar1 = A + (size_t)(m0 + 16 + c) * kdim;
  const float* ar2 = A + (size_t)(m0 + 32 + c) * kdim;
  const float* ar3 = A + (size_t)(m0 + 48 + c) * kdim;
  const float* wr0 = W + (size_t)(n0 +  0 + c) * kdim;
  const float* wr1 = W + (size_t)(n0 + 16 + c) * kdim;
  const float* wr2 = W + (size_t)(n0 + 32 + c) * kdim;
  const float* wr3 = W + (size_t)(n0 + 48 + c) * kdim;
  for (int k0 = 0; k0 < kdim; k0 += 4) {
    const int off = k0 + 2 * hh;
    v2f a[4], b[4];
    a[0] = *(const v2f*)(ar0 + off);
    a[1] = *(const v2f*)(ar1 + off);
    a[2] = *(const v2f*)(ar2 + off);
    a[3] = *(const v2f*)(ar3 + off);
    b[0] = *(const v2f*)(wr0 + off);
    b[1] = *(const v2f*)(wr1 + off);
    b[2] = *(const v2f*)(wr2 + off);
    b[3] = *(const v2f*)(wr3 + off);
#pragma unroll
    for (int mt = 0; mt < 4; ++mt)
#pragma unroll
      for (int nt = 0; nt < 4; ++nt)
        acc[mt][nt] = wmma_f32(a[mt], b[nt], acc[mt][nt]);
  }
}

// ---------------------------------------------------------------------------
// QKV projection: qkv = X @ Wp^T + b; scatter to Q (scaled), K, V^T layouts.
// X[4096,768], Wp[2304,768]. One wave -> 64x64 output tile.
// ---------------------------------------------------------------------------
__global__ __launch_bounds__(256)
void qkv_proj_kernel(const float* __restrict__ X, const float* __restrict__ Wp,
                     const float* __restrict__ bias,
                     float* __restrict__ Q, float* __restrict__ Kd,
                     float* __restrict__ Vt) {
  const int lane = threadIdx.x & 31;
  const int wave = threadIdx.x >> 5;
  const int wg   = blockIdx.x * 8 + wave;
  const int NT   = NQKV / 64;                 // 36
  const int m0   = (wg / NT) * 64;
  const int n0   = (wg % NT) * 64;
  const int c    = lane & 15;
  const int hh   = lane >> 4;

  v8f z = {};
  v8f acc[4][4] = {{z,z,z,z},{z,z,z,z},{z,z,z,z},{z,z,z,z}};
  wave_gemm_64x64(X, Wp, m0, n0, EMBED, c, hh, acc);

  // whole 64-col tile lies in one section (768 = 12*64)
  const int sec = n0 / EMBED;
#pragma unroll
  for (int nt = 0; nt < 4; ++nt) {
    const int n    = n0 + 16 * nt + c;
    const float bn = bias[n];
    const int win  = n - sec * EMBED;
    const int head = win >> 6;
    const int d    = win & 63;
#pragma unroll
    for (int mt = 0; mt < 4; ++mt) {
#pragma unroll
      for (int i = 0; i < 8; ++i) {
        const int row = m0 + 16 * mt + i + 8 * hh;
        const int bb  = row >> 10;              // / SEQ
        const int ll  = row & (SEQ - 1);
        const float v = acc[mt][nt][i] + bn;
        if (sec == 0)
          Q [((size_t)(bb*NH + head)*SEQ + ll)*HD + d] = v * 0.125f; // hd^-0.5
        else if (sec == 1)
          Kd[((size_t)(bb*NH + head)*SEQ + ll)*HD + d] = v;
        else
          Vt[((size_t)(bb*NH + head)*HD + d)*SEQ + ll] = v;
      }
    }
  }
}

// ---------------------------------------------------------------------------
// Fused flash attention per (b,h): scores via WMMA, dual rel-bias, online
// softmax, PV via WMMA (P staged through per-wave LDS for layout swap).
// grid = (B*H, SEQ/(32*8)), block = 256 (8 waves, each a 32-row Q tile).
// Each K/V fragment feeds 2 WMMAs (two 16-row Q subtiles).
// ---------------------------------------------------------------------------
__global__ __launch_bounds__(256)
void attn_kernel(const float* __restrict__ Q, const float* __restrict__ Kd,
                 const float* __restrict__ Vt, const float* __restrict__ relb,
                 const float* __restrict__ patch, const float* __restrict__ dtb,
                 float* __restrict__ attn) {
  __shared__ __align__(16) float s_bias[2*MAXREL + 2];
  __shared__ __align__(16) float s_patch[SEQ];
  __shared__ __align__(16) float s_P[8][32 * 16];

  const int bh = blockIdx.x;
  const int b  = bh / NH;
  const int h  = bh % NH;
  const int tid = threadIdx.x;
  for (int i = tid; i < 2*MAXREL + 1; i += 256) s_bias[i] = relb[h*(2*MAXREL+1) + i];
  for (int i = tid; i < SEQ; i += 256)          s_patch[i] = patch[b*SEQ + i];
  __syncthreads();

  const float dt   = dtb[b];
  const int lane   = tid & 31;
  const int wave   = tid >> 5;
  const int c      = lane & 15;
  const int hh     = lane >> 4;
  const int qbase  = (blockIdx.y * 8 + wave) * 32;

  v2f aq[2][16];
#pragma unroll
  for (int r = 0; r < 2; ++r) {
    const float* qrow = Q + ((size_t)bh * SEQ + qbase + 16*r + c) * HD;
#pragma unroll
    for (int kk = 0; kk < 16; ++kk) aq[r][kk] = *(const v2f*)(qrow + 4*kk + 2*hh);
  }

  float pq[2][8];
#pragma unroll
  for (int r = 0; r < 2; ++r)
#pragma unroll
    for (int i = 0; i < 8; ++i) pq[r][i] = s_patch[qbase + 16*r + i + 8*hh];

  v8f z = {};
  v8f o[2][4] = {{z,z,z,z},{z,z,z,z}};
  float mrow[2][8], lrow[2][8];
#pragma unroll
  for (int r = 0; r < 2; ++r)
#pragma unroll
    for (int i = 0; i < 8; ++i) { mrow[r][i] = -__builtin_inff(); lrow[r][i] = 0.0f; }

  float* myP = &s_P[wave][0];
  const float* vbase = Vt + (size_t)bh * HD * SEQ;

  for (int kb = 0; kb < SEQ; kb += 16) {
    // ---- S[r] = Q_r K^T (two 16x16 tiles sharing each K fragment) ----
    v8f s[2] = {z, z};
    const float* krow = Kd + ((size_t)bh * SEQ + kb + c) * HD;
#pragma unroll
    for (int kk = 0; kk < 16; ++kk) {
      v2f bk = *(const v2f*)(krow + 4*kk + 2*hh);
      s[0] = wmma_f32(aq[0][kk], bk, s[0]);
      s[1] = wmma_f32(aq[1][kk], bk, s[1]);
    }

    // ---- biases + online softmax ----
    const int   kj = kb + c;
    const float pj = s_patch[kj];
#pragma unroll
    for (int r = 0; r < 2; ++r) {
#pragma unroll
      for (int i = 0; i < 8; ++i) {
        const int qi = qbase + 16*r + i + 8*hh;
        int rp = kj - qi;
        rp = rp < -MAXREL ? -MAXREL : (rp > MAXREL ? MAXREL : rp);
        const float b1 = s_bias[rp + MAXREL];
        float rt = pq[r][i] - pj;
        if (__builtin_isnan(rt) || rt < -(float)MAXREL || rt > (float)MAXREL) rt = 0.0f;
        float ri = rintf(rt / dt);                        // half-to-even, like jnp.round
        ri = fminf(fmaxf(ri, -(float)MAXREL), (float)MAXREL);
        const float b2 = s_bias[(int)ri + MAXREL];
        float sv = s[r][i] + b1 + b2;

        float rm = sv;                                    // row max over 16 lanes/half
        rm = fmaxf(rm, __shfl_xor(rm, 1, 32));
        rm = fmaxf(rm, __shfl_xor(rm, 2, 32));
        rm = fmaxf(rm, __shfl_xor(rm, 4, 32));
        rm = fmaxf(rm, __shfl_xor(rm, 8, 32));
        const float mn = fmaxf(mrow[r][i], rm);
        const float al = expf(mrow[r][i] - mn);
        mrow[r][i] = mn;
        const float pv = expf(sv - mn);
        float rs = pv;                                    // row sum
        rs += __shfl_xor(rs, 1, 32);
        rs += __shfl_xor(rs, 2, 32);
        rs += __shfl_xor(rs, 4, 32);
        rs += __shfl_xor(rs, 8, 32);
        lrow[r][i] = lrow[r][i] * al + rs;
        o[r][0][i] *= al; o[r][1][i] *= al; o[r][2][i] *= al; o[r][3][i] *= al;
        myP[(16*r + i + 8*hh) * 16 + c] = pv;             // D-layout -> LDS
      }
    }

    // ---- O += P V; each V fragment feeds both Q subtiles ----
#pragma unroll
    for (int kk = 0; kk < 4; ++kk) {
      const int ko = 4*kk + 2*hh;
      v2f ap0 = *(const v2f*)(myP + ( 0 + c) * 16 + ko);
      v2f ap1 = *(const v2f*)(myP + (16 + c) * 16 + ko);
#pragma unroll
      for (int nt = 0; nt < 4; ++nt) {
        v2f bv = *(const v2f*)(vbase + (size_t)(16*nt + c)*SEQ + kb + ko);
        o[0][nt] = wmma_f32(ap0, bv, o[0][nt]);
        o[1][nt] = wmma_f32(ap1, bv, o[1][nt]);
      }
    }
  }

  // ---- epilogue: normalize and write [B,L,H*hd] ----
#pragma unroll
  for (int r = 0; r < 2; ++r) {
#pragma unroll
    for (int i = 0; i < 8; ++i) {
      const float inv = 1.0f / lrow[r][i];
      const int row = qbase + 16*r + i + 8*hh;
      float* orow = attn + ((size_t)(b*SEQ + row)) * EMBED + h * HD;
      orow[ 0 + c] = o[r][0][i] * inv;
      orow[16 + c] = o[r][1][i] * inv;
      orow[32 + c] = o[r][2][i] * inv;
      orow[48 + c] = o[r][3][i] * inv;
    }
  }
}

// ---------------------------------------------------------------------------
// Output projection: out = attn @ Wout^T + b. Wave -> 64x64 tile, f32 WMMA.
// ---------------------------------------------------------------------------
__global__ __launch_bounds__(256)
void out_proj_kernel(const float* __restrict__ X, const float* __restrict__ Wp,
                     const float* __restrict__ bias, float* __restrict__ Y) {
  const int lane = threadIdx.x & 31;
  const int wave = threadIdx.x >> 5;
  const int wg   = blockIdx.x * 8 + wave;
  const int NT   = EMBED / 64;                // 12
  const int m0   = (wg / NT) * 64;
  const int n0   = (wg % NT) * 64;
  const int c    = lane & 15;
  const int hh   = lane >> 4;

  v8f z = {};
  v8f acc[4][4] = {{z,z,z,z},{z,z,z,z},{z,z,z,z},{z,z,z,z}};
  wave_gemm_64x64(X, Wp, m0, n0, EMBED, c, hh, acc);

#pragma unroll
  for (int nt = 0; nt < 4; ++nt) {
    const int n = n0 + 16 * nt + c;
    const float bn = bias[n];
#pragma unroll
    for (int mt = 0; mt < 4; ++mt) {
#pragma unroll
      for (int i = 0; i < 8; ++i) {
        const int row = m0 + 16 * mt + i + 8 * hh;
        Y[(size_t)row * EMBED + n] = acc[mt][nt][i] + bn;
      }
    }
  }
}

// ---------------------------------------------------------------------------
extern "C" void kernel_launch(void* const* d_in, const int* in_sizes, int n_in,
                              void* d_out, int out_size, void* d_ws, size_t ws_size,
                              hipStream_t stream) {
  const float* query = (const float*)d_in[0];
  // d_in[1]=key, d_in[2]=value: ignored (reference builds qkv from query)
  const float* patch = (const float*)d_in[3];
  const float* in_w  = (const float*)d_in[4];
  const float* in_b  = (const float*)d_in[5];
  const float* out_w = (const float*)d_in[6];
  const float* out_b = (const float*)d_in[7];
  const float* relb  = (const float*)d_in[8];
  float* out = (float*)d_out;

  float* wsf = (float*)d_ws;
  const size_t SZ = (size_t)NB * NH * SEQ * HD;   // 3,145,728 floats per tensor
  float* dt   = wsf;
  float* Q    = wsf + 256;
  float* Kd   = Q  + SZ;
  float* Vt   = Kd + SZ;
  float* attn = Vt + SZ;

  median_dt_kernel<<<NB, 1024, 0, stream>>>(patch, dt);

  // (4096/64) * (2304/64) = 2304 wave-tiles / 8 waves per block = 288 blocks
  qkv_proj_kernel<<<288, 256, 0, stream>>>(query, in_w, in_b, Q, Kd, Vt);

  attn_kernel<<<dim3(NB * NH, SEQ / (32 * 8)), 256, 0, stream>>>(
      Q, Kd, Vt, relb, patch, dt, attn);

  // (4096/64) * (768/64) = 768 wave-tiles / 8 waves per block = 96 blocks
  out_proj_kernel<<<96, 256, 0, stream>>>(attn, out_w, out_b, out);
}